// Decoder_30846455120266
// MI455X (gfx1250) — compile-verified
//
#include <hip/hip_runtime.h>
#include <hip/hip_bf16.h>

// Problem constants (match reference)
#define H_    512
#define H2_   1024
#define H4_   2048
#define V_    10000
#define B_    32
#define T_    1000
#define L_    100
#define TCH_  8                    // t-chunks for context reduction
#define INF_MIN_ (-1.0e10f)

typedef __attribute__((ext_vector_type(2))) float v2f;
typedef __attribute__((ext_vector_type(4))) float v4f;
typedef __attribute__((ext_vector_type(8))) float v8f;

__device__ __forceinline__ float fast_tanh(float x) {
  float cx = fminf(fmaxf(x, -15.0f), 15.0f);
  float e2 = __expf(2.0f * cx);          // v_exp_f32 path
  return (e2 - 1.0f) / (e2 + 1.0f);
}
__device__ __forceinline__ float ref_sigmoid(float x) {
  // reference: tanh(x*0.5)*0.5 + 0.5
  return 0.5f * fast_tanh(0.5f * x) + 0.5f;
}

// ---------------------------------------------------------------------------
// Generic f32 WMMA GEMM (double-buffered, software-pipelined staging):
//   C[m0:m0+32, n0:n0+64] = act( A1@W1 (+ A2@W2) + bias1 + bias2 + Eyr[tok[m]] )
// A row-major [M x K] (lda), W row-major [K x N] (ld = N), C row-major (ldc).
// Block: 256 threads = 8 waves; wave w -> one 16x16 tile (wm = w&1, wn = w>>1).
// WMMA f32 16x16x4 operand layouts (CDNA5 ISA 7.12.2):
//   A[j] (lane) = A[m0 + lane%16][kb + j + 2*(lane/16)]
//   B[j] (lane) = W[kb + j + 2*(lane/16)][n0 + lane%16]
//   C[r] (lane) = C[m0 + r + 8*(lane/16)][n0 + lane%16]
// Staging is static & 128b-vectorized:
//   A tile 32x32: thread -> row tid/8, float4 at col 4*(tid%8)
//   W tile 32x64: thread -> row tid/8, two float4 at col 8*(tid%8)
// Pipeline: preload chunk i+1 into regs right after the barrier, compute chunk
// i from LDS buffer buf, store the regs into the other LDS buffer next iteration (one barrier
// per chunk; global-load latency overlapped with WMMA work).
// LDS strides: SAS=36 (rows 16B aligned, frag reads conflict-free),
//              SBS=72 (hi-half lanes +16 banks from lo-half -> disjoint)
// ---------------------------------------------------------------------------
#define KC  32
#define BN  64
#define SAS 36
#define SBS 72

__global__ __launch_bounds__(256) void wmma_gemm_f32(
    const float* __restrict__ A1, int lda1, int K1, const float* __restrict__ W1,
    const float* __restrict__ A2, int lda2, int K2, const float* __restrict__ W2,
    const float* __restrict__ bias1, const float* __restrict__ bias2,
    const float* __restrict__ Eyr, const int* __restrict__ tok, int tokStride,
    float* __restrict__ C, int ldc, int N, int act)
{
  __shared__ __align__(16) float sA[2 * 32 * SAS];
  __shared__ __align__(16) float sB[2 * KC * SBS];

  const int m0   = blockIdx.y * 32;
  const int n0   = blockIdx.x * BN;
  const int tid  = threadIdx.x;
  const int w    = tid >> 5;
  const int lane = tid & 31;
  const int lo   = lane & 15;
  const int hi   = lane >> 4;
  const int wm   = w & 1;
  const int wn   = w >> 1;

  // static staging assignment
  const int srow = tid >> 3;            // 0..31
  const int sq   = tid & 7;             // 0..7
  const int wcol = n0 + 8 * sq;         // W cols (two float4); N % 8 == 0
  const bool wok = (wcol < N);

  // fragment read bases
  const int arow = (16 * wm + lo) * SAS;
  const int bcol = 16 * wn + lo;

  v8f acc0 = {};
  v8f acc1 = {};

  const int nch1 = K1 / KC;
  const int nch2 = (A2 != nullptr) ? (K2 / KC) : 0;
  const int nch  = nch1 + nch2;
  const size_t wStep = (size_t)KC * N;

  // preload chunk 0
  const float* aPtr = A1 + (size_t)(m0 + srow) * lda1 + 4 * sq;
  const float* wPtr = W1 + (size_t)srow * N + wcol;
  v4f aReg = *(const v4f*)aPtr;
  v4f wReg0 = {};
  v4f wReg1 = {};
  if (wok) {
    wReg0 = *(const v4f*)wPtr;
    wReg1 = *(const v4f*)(wPtr + 4);
  }

  int buf = 0;
  for (int i = 0; i < nch; ++i) {
    float* dA = sA + buf * (32 * SAS);
    float* dB = sB + buf * (KC * SBS);
    // commit prefetched chunk i to LDS
    *(v4f*)&dA[srow * SAS + 4 * sq]     = aReg;
    *(v4f*)&dB[srow * SBS + 8 * sq]     = wReg0;
    *(v4f*)&dB[srow * SBS + 8 * sq + 4] = wReg1;
    __syncthreads();

    // issue global loads for chunk i+1 (overlap with WMMA below)
    if (i + 1 < nch) {
      if (i + 1 == nch1) {
        aPtr = A2 + (size_t)(m0 + srow) * lda2 + 4 * sq;
        wPtr = W2 + (size_t)srow * N + wcol;
      } else {
        aPtr += KC;
        wPtr += wStep;
      }
      aReg = *(const v4f*)aPtr;
      if (wok) {
        wReg0 = *(const v4f*)wPtr;
        wReg1 = *(const v4f*)(wPtr + 4);
      }
    }

    // compute chunk i from LDS
#pragma unroll
    for (int ks = 0; ks < KC / 4; ++ks) {
      const int ka = 4 * ks + 2 * hi;
      v2f a, b;
      a.x = dA[arow + ka];
      a.y = dA[arow + ka + 1];
      b.x = dB[ka * SBS + bcol];
      b.y = dB[(ka + 1) * SBS + bcol];
      if (ks & 1)
        acc1 = __builtin_amdgcn_wmma_f32_16x16x4_f32(
            false, a, false, b, (short)0, acc1, false, false);
      else
        acc0 = __builtin_amdgcn_wmma_f32_16x16x4_f32(
            false, a, false, b, (short)0, acc0, false, false);
    }
    buf ^= 1;
  }

  // Epilogue
  const int col = n0 + 16 * wn + lo;
  if (col < N) {
    float badd = 0.0f;
    if (bias1) badd += bias1[col];
    if (bias2) badd += bias2[col];
#pragma unroll
    for (int r = 0; r < 8; ++r) {
      const int row = m0 + 16 * wm + 8 * hi + r;
      float v = acc0[r] + acc1[r] + badd;
      if (Eyr) v += Eyr[(size_t)tok[row * tokStride] * N + col];
      if (act) v = fast_tanh(v);
      C[(size_t)row * ldc + col] = v;
    }
  }
}

// ---------------------------------------------------------------------------
// e[b,t] = masked( sum_a w_att[a] * tanh(h_proj[b,t,a] + sW[b,a]) )
// One wave per timestep; lanes split H=512 as float4 (4 iterations).
// ---------------------------------------------------------------------------
__global__ __launch_bounds__(256) void attn_energy_kernel(
    const float* __restrict__ hproj, const float* __restrict__ sW,
    const float* __restrict__ w_att, const int* __restrict__ seq_lens,
    float* __restrict__ e)
{
  const int b = blockIdx.y;
  __shared__ __align__(16) float sw[H_];
  __shared__ __align__(16) float wa[H_];
  for (int i = threadIdx.x; i < H_ / 4; i += 256) {
    ((v4f*)sw)[i] = ((const v4f*)(sW + b * H_))[i];
    ((v4f*)wa)[i] = ((const v4f*)w_att)[i];
  }
  __syncthreads();
  const int wv   = threadIdx.x >> 5;
  const int lane = threadIdx.x & 31;
  const int t    = blockIdx.x * 8 + wv;              // grid.x*8 == T_ exactly
  const float* hp = hproj + ((size_t)b * T_ + t) * H_;
  float acc = 0.0f;
#pragma unroll
  for (int it = 0; it < 4; ++it) {
    const int a = 128 * it + 4 * lane;
    const v4f h  = *(const v4f*)(hp + a);
    const v4f s4 = *(const v4f*)(sw + a);
    const v4f w4 = *(const v4f*)(wa + a);
    acc += fast_tanh(h.x + s4.x) * w4.x;
    acc += fast_tanh(h.y + s4.y) * w4.y;
    acc += fast_tanh(h.z + s4.z) * w4.z;
    acc += fast_tanh(h.w + s4.w) * w4.w;
  }
#pragma unroll
  for (int off = 16; off; off >>= 1) acc += __shfl_xor(acc, off);
  if (lane == 0)
    e[b * T_ + t] = (t < seq_lens[b]) ? acc : INF_MIN_;
}

// ---------------------------------------------------------------------------
// Row softmax over T per batch (one 1024-thread block per b).
// ---------------------------------------------------------------------------
__global__ __launch_bounds__(1024) void softmax_rows(
    const float* __restrict__ e, float* __restrict__ alpha)
{
  const int b    = blockIdx.x;
  const int tid  = threadIdx.x;
  const int lane = tid & 31;
  const int wid  = tid >> 5;
  __shared__ float redm[32];
  __shared__ float reds[32];
  __shared__ float sM, sS;

  float v = (tid < T_) ? e[(size_t)b * T_ + tid] : -3.0e38f;
  float m = v;
#pragma unroll
  for (int off = 16; off; off >>= 1) m = fmaxf(m, __shfl_xor(m, off));
  if (lane == 0) redm[wid] = m;
  __syncthreads();
  if (wid == 0) {
    float mm = redm[lane];
#pragma unroll
    for (int off = 16; off; off >>= 1) mm = fmaxf(mm, __shfl_xor(mm, off));
    if (lane == 0) sM = mm;
  }
  __syncthreads();
  const float ev = (tid < T_) ? __expf(v - sM) : 0.0f;
  float s = ev;
#pragma unroll
  for (int off = 16; off; off >>= 1) s += __shfl_xor(s, off);
  if (lane == 0) reds[wid] = s;
  __syncthreads();
  if (wid == 0) {
    float ss = reds[lane];
#pragma unroll
    for (int off = 16; off; off >>= 1) ss += __shfl_xor(ss, off);
    if (lane == 0) sS = ss;
  }
  __syncthreads();
  if (tid < T_) alpha[(size_t)b * T_ + tid] = ev / sS;
}

// ---------------------------------------------------------------------------
// gpart[ch,b,h] = sum_{t in chunk ch} alpha[b,t] * h_batch[b,t,h]
// grid (TCH_, B_), 256 threads; thread -> float4 of h (4*tid .. 4*tid+3).
// ---------------------------------------------------------------------------
__global__ __launch_bounds__(256) void attn_context_kernel(
    const float* __restrict__ alpha, const float* __restrict__ hb,
    float* __restrict__ gpart)
{
  const int b  = blockIdx.y;
  const int ch = blockIdx.x;
  const int t0 = ch * (T_ / TCH_);
  __shared__ float al[T_ / TCH_];
  for (int i = threadIdx.x; i < T_ / TCH_; i += 256)
    al[i] = alpha[b * T_ + t0 + i];
  __syncthreads();
  const float* base = hb + ((size_t)b * T_ + t0) * H2_ + 4 * threadIdx.x;
  v4f acc = {};
#pragma unroll 4
  for (int t = 0; t < T_ / TCH_; ++t) {
    __builtin_prefetch(base + (size_t)(t + 8) * H2_, 0, 0);  // speculative
    const v4f v = *(const v4f*)(base + (size_t)t * H2_);
    const float a = al[t];
    acc.x += a * v.x;
    acc.y += a * v.y;
    acc.z += a * v.z;
    acc.w += a * v.w;
  }
  *(v4f*)&gpart[((size_t)ch * B_ + b) * H2_ + 4 * threadIdx.x] = acc;
}

__global__ __launch_bounds__(256) void context_reduce_kernel(
    const float* __restrict__ gpart, float* __restrict__ g)
{
  const int i = blockIdx.x * 256 + threadIdx.x;   // < B_*H2_
  float acc = 0.0f;
#pragma unroll
  for (int ch = 0; ch < TCH_; ++ch)
    acc += gpart[(size_t)ch * B_ * H2_ + i];
  g[i] = acc;
}

// ---------------------------------------------------------------------------
// LSTM pointwise update: rec[b, 4H] -> s[b,H], c[b,H]
// ---------------------------------------------------------------------------
__global__ __launch_bounds__(256) void lstm_update_kernel(
    const float* __restrict__ rec, float* __restrict__ s, float* __restrict__ c)
{
  const int idx = blockIdx.x * 256 + threadIdx.x;
  if (idx >= B_ * H_) return;
  const int b = idx >> 9;          // /H_
  const int h = idx & (H_ - 1);
  const float* r = rec + (size_t)b * H4_;
  const float ig = ref_sigmoid(r[h]);
  const float fg = ref_sigmoid(r[H_ + h]);
  const float cg = fast_tanh(r[2 * H_ + h]);
  const float og = ref_sigmoid(r[3 * H_ + h]);
  const float cn = fg * c[idx] + ig * cg;
  c[idx] = cn;
  s[idx] = og * fast_tanh(cn);
}

__global__ void zero_kernel(float* __restrict__ p, int n) {
  const int i = blockIdx.x * blockDim.x + threadIdx.x;
  if (i < n) p[i] = 0.0f;
}

// ---------------------------------------------------------------------------
extern "C" void kernel_launch(void* const* d_in, const int* in_sizes, int n_in,
                              void* d_out, int out_size, void* d_ws, size_t ws_size,
                              hipStream_t stream) {
  (void)in_sizes; (void)n_in; (void)out_size; (void)ws_size;
  const float* h_batch = (const float*)d_in[0];
  const int*   seq_lens= (const int*)  d_in[1];
  const int*   labels  = (const int*)  d_in[2];
  const float* W_sy    = (const float*)d_in[3];
  const float* b_sy    = (const float*)d_in[4];
  const float* W_gy    = (const float*)d_in[5];
  const float* b_gy    = (const float*)d_in[6];
  const float* W_yy    = (const float*)d_in[7];
  const float* b_yy    = (const float*)d_in[8];
  const float* E_yr    = (const float*)d_in[9];
  const float* W_sr    = (const float*)d_in[10];
  const float* b_sr    = (const float*)d_in[11];
  const float* W_gr    = (const float*)d_in[12];
  const float* b_gr    = (const float*)d_in[13];
  const float* W_s_att = (const float*)d_in[14];
  const float* W_h_att = (const float*)d_in[15];
  const float* b_att   = (const float*)d_in[16];
  const float* w_att   = (const float*)d_in[17];
  float* out = (float*)d_out;

  // workspace layout
  float* ws    = (float*)d_ws;
  float* hproj = ws; ws += (size_t)B_ * T_ * H_;   // 16,384,000
  float* sW    = ws; ws += B_ * H_;
  float* e     = ws; ws += B_ * T_;
  float* alpha = ws; ws += B_ * T_;
  float* gpart = ws; ws += (size_t)TCH_ * B_ * H2_;
  float* g     = ws; ws += B_ * H2_;
  float* z     = ws; ws += B_ * H_;
  float* rec   = ws; ws += B_ * H4_;
  float* s     = ws; ws += B_ * H_;
  float* cbuf  = ws; ws += B_ * H_;                // contiguous with s

  // s = c = 0 (deterministic per call; workspace is poisoned by harness)
  zero_kernel<<<(2 * B_ * H_ + 255) / 256, 256, 0, stream>>>(s, 2 * B_ * H_);

  // h_proj = h_batch @ W_h_att + b_att   [B*T, H]
  wmma_gemm_f32<<<dim3(H_ / BN, (B_ * T_) / 32), 256, 0, stream>>>(
      h_batch, H2_, H2_, W_h_att,
      nullptr, 0, 0, nullptr,
      b_att, nullptr,
      nullptr, nullptr, 0,
      hproj, H_, H_, /*act=*/0);

  for (int l = 0; l < L_; ++l) {
    // sW = s @ W_s_att
    wmma_gemm_f32<<<dim3(H_ / BN, 1), 256, 0, stream>>>(
        s, H_, H_, W_s_att, nullptr, 0, 0, nullptr,
        nullptr, nullptr, nullptr, nullptr, 0,
        sW, H_, H_, 0);

    attn_energy_kernel<<<dim3(T_ / 8, B_), 256, 0, stream>>>(
        hproj, sW, w_att, seq_lens, e);

    softmax_rows<<<B_, 1024, 0, stream>>>(e, alpha);

    attn_context_kernel<<<dim3(TCH_, B_), 256, 0, stream>>>(alpha, h_batch, gpart);
    context_reduce_kernel<<<(B_ * H2_) / 256, 256, 0, stream>>>(gpart, g);

    // z = tanh(s@W_sy + g@W_gy + b_sy + b_gy)
    wmma_gemm_f32<<<dim3(H_ / BN, 1), 256, 0, stream>>>(
        s, H_, H_, W_sy, g, H2_, H2_, W_gy,
        b_sy, b_gy, nullptr, nullptr, 0,
        z, H_, H_, /*act=*/1);

    // out[:, l, :] = z @ W_yy + b_yy     (ldc = L*V row stride)
    wmma_gemm_f32<<<dim3((V_ + BN - 1) / BN, 1), 256, 0, stream>>>(
        z, H_, H_, W_yy, nullptr, 0, 0, nullptr,
        b_yy, nullptr, nullptr, nullptr, 0,
        out + (size_t)l * V_, L_ * V_, V_, 0);

    // rec = E_yr[labels[:,l]] + s@W_sr + g@W_gr + b_sr + b_gr
    wmma_gemm_f32<<<dim3(H4_ / BN, 1), 256, 0, stream>>>(
        s, H_, H_, W_sr, g, H2_, H2_, W_gr,
        b_sr, b_gr, E_yr, labels + l, L_,
        rec, H4_, H4_, 0);

    lstm_update_kernel<<<(B_ * H_ + 255) / 256, 256, 0, stream>>>(rec, s, cbuf);
  }
}